// GCN_11158325035600
// MI455X (gfx1250) — compile-verified
//
#include <hip/hip_runtime.h>

typedef float v2f __attribute__((ext_vector_type(2)));
typedef float v8f __attribute__((ext_vector_type(8)));

#define N_NODES 50000
#define N_EDGES 800000
#define N_GRAPHS 50
#define NPG 1000
#define DIM 128
#define OUTD 10
#define N_CONV 4
#define EPS 1e-6f
#define MT_PER_BLK 5            // M-tiles (of 16 rows) per GEMM block; 50000/80 = 625

// ---------------- utility kernels ----------------

__global__ void zero_f4(float4* __restrict__ p, int n4) {
    int i = blockIdx.x * blockDim.x + threadIdx.x;
    if (i < n4) p[i] = make_float4(0.f, 0.f, 0.f, 0.f);
}

__global__ void degree_k(const int* __restrict__ src, const int* __restrict__ dst,
                         float* __restrict__ deg_src, float* __restrict__ deg_dst) {
    int e = blockIdx.x * blockDim.x + threadIdx.x;
    if (e < N_EDGES) {
        atomicAdd(&deg_src[src[e]], 1.0f);
        atomicAdd(&deg_dst[dst[e]], 1.0f);
    }
}

__global__ void invsqrt_k(float* __restrict__ p, int n) {
    int i = blockIdx.x * blockDim.x + threadIdx.x;
    if (i < n) p[i] = rsqrtf(fmaxf(p[i], 1.0f));
}

// one wave per edge: lane c handles floats [4c..4c+3] of the 128-dim message.
// Fuses the source normalization: agg[dst] += h[src] * norm_src[src]
__global__ void scatter_k(const int* __restrict__ src, const int* __restrict__ dst,
                          const float4* __restrict__ h, const float* __restrict__ ns,
                          float* __restrict__ agg) {
    long long i = (long long)blockIdx.x * blockDim.x + threadIdx.x;
    int e = (int)(i >> 5);
    int c = (int)(i & 31);
    if (e < N_EDGES) {
        int s = src[e];
        int d = dst[e];
        float  sc = ns[s];                      // uniform across the wave (broadcast)
        float4 v = h[s * (DIM / 4) + c];
        float* a = agg + d * DIM + c * 4;
        atomicAdd(a + 0, v.x * sc);
        atomicAdd(a + 1, v.y * sc);
        atomicAdd(a + 2, v.z * sc);
        atomicAdd(a + 3, v.w * sc);
    }
}

// ---------------- WMMA GEMM: hc = (agg * norm_dst[row]) @ W + bias ----------------
// grid.x = N_NODES/(16*MT_PER_BLK), block = 256 (8 waves), wave w -> N tile w.
// Each block stages the full 128x128 W in LDS once, then computes MT_PER_BLK
// 16-row M tiles: per K-step one B fetch feeds MT_PER_BLK back-to-back WMMAs.
__global__ __launch_bounds__(256)
void gemm_wmma_k(const float* __restrict__ agg, const float* __restrict__ nd,
                 const float* __restrict__ W, const float* __restrict__ bias,
                 float* __restrict__ hc) {
    __shared__ float Bsh[DIM * DIM];            // 64 KB LDS (of 320 KB/WGP)

    {
        const float4* W4 = (const float4*)W;
        float4* B4 = (float4*)Bsh;
        for (int i = threadIdx.x; i < DIM * DIM / 4; i += 256) B4[i] = W4[i];
    }
    __syncthreads();

    const int wave = threadIdx.x >> 5;
    const int lane = threadIdx.x & 31;
    const int nBase = wave * 16;
    const int mn   = lane & 15;                 // A: M index / B: N index (ISA 16x4/4x16 layout)
    const int half = lane >> 4;                 // lane-half selects the K pair

    const int mBase0 = blockIdx.x * (16 * MT_PER_BLK);

    const float* Arow[MT_PER_BLK];
    float scale[MT_PER_BLK];
    v8f c[MT_PER_BLK];
#pragma unroll
    for (int mt = 0; mt < MT_PER_BLK; ++mt) {
        int row = mBase0 + 16 * mt + mn;
        Arow[mt] = agg + row * DIM + 2 * half;
        scale[mt] = nd[row];
        c[mt] = (v8f){};
    }

#pragma unroll 2
    for (int t = 0; t < DIM / 4; ++t) {
        // B 4x16 f32 layout: lane n, the two VGPRs hold rows K=2*half, 2*half+1
        int k = 4 * t + 2 * half;
        v2f b;
        b.x = Bsh[k * DIM + nBase + mn];
        b.y = Bsh[(k + 1) * DIM + nBase + mn];
#pragma unroll
        for (int mt = 0; mt < MT_PER_BLK; ++mt) {
            // A 16x4 f32 layout: lanes 0-15 -> K={0,1}, lanes 16-31 -> K={2,3}
            float2 av = *(const float2*)(Arow[mt] + 4 * t);
            v2f a;
            a.x = av.x * scale[mt];
            a.y = av.y * scale[mt];
            c[mt] = __builtin_amdgcn_wmma_f32_16x16x4_f32(false, a, false, b,
                                                          (short)0, c[mt], false, false);
        }
    }

    const float bb = bias[nBase + mn];
#pragma unroll
    for (int mt = 0; mt < MT_PER_BLK; ++mt) {
        // D layout: VGPR r -> row r + 8*half, col = lane&15
#pragma unroll
        for (int r = 0; r < 8; ++r) {
            int row = mBase0 + 16 * mt + r + 8 * half;
            hc[row * DIM + nBase + mn] = c[mt][r] + bb;
        }
    }
}

// ---------------- GraphNorm + ReLU + residual (in place) ----------------
// grid = N_GRAPHS, block = DIM; thread d owns feature dim d of graph g
__global__ void graphnorm_k(float* __restrict__ hc, const float* __restrict__ xres,
                            const float* __restrict__ gw, const float* __restrict__ gb,
                            const float* __restrict__ gms, int add_res) {
    const int g = blockIdx.x;
    const int d = threadIdx.x;
    const int base = g * NPG;

    float s = 0.f;
    for (int n = 0; n < NPG; ++n) s += hc[(base + n) * DIM + d];
    const float mean = s * (1.0f / NPG);
    const float shift = mean * gms[d];

    float v = 0.f;
    for (int n = 0; n < NPG; ++n) {
        float sub = hc[(base + n) * DIM + d] - shift;
        v += sub * sub;
    }
    const float inv = rsqrtf(v * (1.0f / NPG) + EPS);
    const float w = gw[d], b = gb[d];

    for (int n = 0; n < NPG; ++n) {
        int idx = (base + n) * DIM + d;
        float sub = hc[idx] - shift;
        float y = fmaxf(w * sub * inv + b, 0.0f);
        if (add_res) y += xres[idx];
        hc[idx] = y;
    }
}

// ---------------- sum pool + output projection ----------------
// grid = N_GRAPHS, block = DIM
__global__ void pool_out_k(const float* __restrict__ h, const float* __restrict__ Wo,
                           const float* __restrict__ bo, float* __restrict__ out) {
    __shared__ float pooled[DIM];
    const int g = blockIdx.x;
    const int d = threadIdx.x;
    float s = 0.f;
    for (int n = 0; n < NPG; ++n) s += h[(g * NPG + n) * DIM + d];
    pooled[d] = s;
    __syncthreads();
    if (d < OUTD) {
        float acc = bo[d];
        for (int k = 0; k < DIM; ++k) acc += pooled[k] * Wo[k * OUTD + d];
        out[g * OUTD + d] = acc;
    }
}

// ---------------- launcher ----------------

extern "C" void kernel_launch(void* const* d_in, const int* in_sizes, int n_in,
                              void* d_out, int out_size, void* d_ws, size_t ws_size,
                              hipStream_t stream) {
    (void)in_sizes; (void)n_in; (void)out_size; (void)ws_size;

    const float* h    = (const float*)d_in[0];   // [N_NODES, DIM]
    const float* Wc   = (const float*)d_in[1];   // [N_CONV, DIM, DIM]
    const float* bc   = (const float*)d_in[2];   // [N_CONV, DIM]
    const float* nw   = (const float*)d_in[3];   // [N_CONV, DIM]
    const float* nb   = (const float*)d_in[4];   // [N_CONV, DIM]
    const float* nms  = (const float*)d_in[5];   // [N_CONV, DIM]
    const float* Wo   = (const float*)d_in[6];   // [DIM, OUTD]
    const float* bo   = (const float*)d_in[7];   // [OUTD]
    const int*   esrc = (const int*)d_in[8];     // [N_EDGES]
    const int*   edst = (const int*)d_in[9];     // [N_EDGES]
    float* out = (float*)d_out;                  // [N_GRAPHS, OUTD]

    char* ws = (char*)d_ws;
    float* norm_src = (float*)ws; ws += (size_t)N_NODES * 4;          // also deg_out scratch
    float* norm_dst = (float*)ws; ws += (size_t)N_NODES * 4;          // also deg_in scratch
    float* agg  = (float*)ws;     ws += (size_t)N_NODES * DIM * 4;
    float* bufA = (float*)ws;     ws += (size_t)N_NODES * DIM * 4;
    float* bufB = (float*)ws;

    // ---- degrees -> norms (norm_src/norm_dst contiguous: zero & rsqrt in one pass) ----
    zero_f4<<<(2 * N_NODES / 4 + 255) / 256, 256, 0, stream>>>((float4*)norm_src, 2 * N_NODES / 4);
    degree_k<<<(N_EDGES + 255) / 256, 256, 0, stream>>>(esrc, edst, norm_src, norm_dst);
    invsqrt_k<<<(2 * N_NODES + 255) / 256, 256, 0, stream>>>(norm_src, 2 * N_NODES);

    const int node_chunks = N_NODES * (DIM / 4);      // 1.6M float4s
    const long long edge_threads = (long long)N_EDGES * 32;

    const float* hcur = h;
    float* bufs[2] = {bufA, bufB};
    for (int i = 0; i < N_CONV; ++i) {
        float* hout = bufs[i & 1];
        zero_f4<<<(node_chunks + 255) / 256, 256, 0, stream>>>((float4*)agg, node_chunks);
        scatter_k<<<(unsigned)((edge_threads + 255) / 256), 256, 0, stream>>>(
            esrc, edst, (const float4*)hcur, norm_src, agg);
        gemm_wmma_k<<<N_NODES / (16 * MT_PER_BLK), 256, 0, stream>>>(
            agg, norm_dst, Wc + (size_t)i * DIM * DIM, bc + (size_t)i * DIM, hout);
        graphnorm_k<<<N_GRAPHS, DIM, 0, stream>>>(
            hout, hcur, nw + (size_t)i * DIM, nb + (size_t)i * DIM, nms + (size_t)i * DIM,
            i != 0 ? 1 : 0);
        hcur = hout;
    }

    pool_out_k<<<N_GRAPHS, DIM, 0, stream>>>(hcur, Wo, bo, out);
}